// Model21_3015067041928
// MI455X (gfx1250) — compile-verified
//
#include <hip/hip_runtime.h>

// Problem constants (from reference): xyz [4,16384,3], points [4,16384,64]
#define BATCH   4
#define NPTS    16384
#define DCH     64
#define SGRP    2048      // GROUPS
#define KNN     32        // KNEIGHBORS
#define CHUNKS  8
#define CHUNK_N (NPTS / CHUNKS)          // 2048
#define CT      16                        // candidate tile (WMMA N)
#define TILES_PER_CHUNK (CHUNK_N / CT)    // 128
#define QPW     32                        // queries per wave (two 16-row A tiles)
#define PPT     16                        // points per thread in FPS (16384/1024)

// d_out float offsets: new_xyz | new_points | grouped_xyz | grouped_points
#define O_NEWPTS (BATCH * SGRP * 3)                       // 24576
#define O_GXYZ   (O_NEWPTS + BATCH * SGRP * DCH)          // 548864
#define O_GPTS   (O_GXYZ + BATCH * SGRP * KNN * 3)        // 1335296

typedef __attribute__((ext_vector_type(2))) float v2f;
typedef __attribute__((ext_vector_type(8))) float v8f;

// ---------------------------------------------------------------------------
// Kernel 1: pack candidates as (-2x, -2y, -2z, |q|^2)  -> one float4 / point
// ---------------------------------------------------------------------------
__global__ void prep_cand(const float* __restrict__ xyz, float* __restrict__ cand) {
    int g = blockIdx.x * blockDim.x + threadIdx.x;   // b*NPTS + n
    if (g >= BATCH * NPTS) return;
    float x = xyz[(size_t)g * 3 + 0];
    float y = xyz[(size_t)g * 3 + 1];
    float z = xyz[(size_t)g * 3 + 2];
    float4 r;
    r.x = -2.0f * x; r.y = -2.0f * y; r.z = -2.0f * z;
    r.w = x * x + y * y + z * z;
    ((float4*)cand)[g] = r;
}

// ---------------------------------------------------------------------------
// Kernel 2: farthest point sampling. One workgroup per batch. Per-thread
// point coordinates AND running min-distances live in registers (the 2048
// dependent iterations are the serial critical path; register residency
// removes ~48 LDS loads/thread/iter). LDS keeps a copy of the cloud only for
// the broadcast centroid fetch (3 ds_load_b32 per iteration, same-address
// broadcast), exploiting CDNA5's 320KB/WGP LDS (3*64KB = 192KB).
// 2 barriers per iteration.
// ---------------------------------------------------------------------------
__global__ __launch_bounds__(1024) void fps_kernel(const float* __restrict__ xyz,
                                                   int* __restrict__ fpsIdx) {
    __shared__ float sx[NPTS];
    __shared__ float sy[NPTS];
    __shared__ float sz[NPTS];
    __shared__ float redV[32];
    __shared__ int   redI[32];
    __shared__ int   farSh;

    const int b = blockIdx.x;
    const int t = threadIdx.x;
    const float* base = xyz + (size_t)b * NPTS * 3;

    for (int i = t; i < NPTS; i += 1024) {
        sx[i] = base[(size_t)i * 3 + 0];
        sy[i] = base[(size_t)i * 3 + 1];
        sz[i] = base[(size_t)i * 3 + 2];
    }
    // register-resident private points + running distances
    float px[PPT], py[PPT], pz[PPT], dist[PPT];
#pragma unroll
    for (int j = 0; j < PPT; ++j) {
        const int p = t + j * 1024;
        px[j] = base[(size_t)p * 3 + 0];
        py[j] = base[(size_t)p * 3 + 1];
        pz[j] = base[(size_t)p * 3 + 2];
        dist[j] = 1e10f;
    }
    if (t == 0) farSh = 0;
    __syncthreads();

    int far = 0;
    const int lane = t & 31;
    const int w    = t >> 5;

    for (int it = 0; it < SGRP; ++it) {
        if (t == 0) fpsIdx[b * SGRP + it] = far;   // emit pre-update (ref scan order)
        const float cx = sx[far], cy = sy[far], cz = sz[far];  // LDS broadcast

        float bv = -1.0f; int bi = 0;
#pragma unroll
        for (int j = 0; j < PPT; ++j) {
            const float dx = px[j] - cx, dy = py[j] - cy, dz = pz[j] - cz;
            const float d  = dx * dx + dy * dy + dz * dz;
            const float nd = fminf(dist[j], d);
            dist[j] = nd;
            if (nd > bv) { bv = nd; bi = t + j * 1024; }  // ascending p -> first max
        }
        // wave32 argmax reduce (tie -> lower index, matching jnp.argmax)
        float v = bv; int i = bi;
#pragma unroll
        for (int off = 16; off; off >>= 1) {
            float ov = __shfl_xor(v, off, 32);
            int   oi = __shfl_xor(i, off, 32);
            if (ov > v || (ov == v && oi < i)) { v = ov; i = oi; }
        }
        if (lane == 0) { redV[w] = v; redI[w] = i; }
        __syncthreads();
        if (w == 0) {
            float v2 = redV[lane]; int i2 = redI[lane];
#pragma unroll
            for (int off = 16; off; off >>= 1) {
                float ov = __shfl_xor(v2, off, 32);
                int   oi = __shfl_xor(i2, off, 32);
                if (ov > v2 || (ov == v2 && oi < i2)) { v2 = ov; i2 = oi; }
            }
            if (lane == 0) farSh = i2;
        }
        __syncthreads();
        far = farSh;
    }
}

// ---------------------------------------------------------------------------
// Kernel 3: gather sampled centroids -> new_xyz, new_points, and packed
// query records (x, y, z, |p|^2) for the WMMA distance kernel.
// ---------------------------------------------------------------------------
__global__ void qgather(const float* __restrict__ xyz, const float* __restrict__ pts,
                        const int* __restrict__ fpsIdx,
                        float* __restrict__ out, float* __restrict__ qpack) {
    int g = blockIdx.x * blockDim.x + threadIdx.x;   // b*SGRP + s
    if (g >= BATCH * SGRP) return;
    const int b = g / SGRP;
    const int idx = fpsIdx[g];
    const float* xr = xyz + ((size_t)b * NPTS + idx) * 3;
    const float x = xr[0], y = xr[1], z = xr[2];
    out[(size_t)g * 3 + 0] = x;
    out[(size_t)g * 3 + 1] = y;
    out[(size_t)g * 3 + 2] = z;
    float4 q; q.x = x; q.y = y; q.z = z; q.w = x * x + y * y + z * z;
    ((float4*)qpack)[g] = q;
    const float4* pr = (const float4*)(pts + ((size_t)b * NPTS + idx) * DCH);
    float4* po = (float4*)(out + O_NEWPTS + (size_t)g * DCH);
#pragma unroll
    for (int c = 0; c < DCH / 4; ++c) po[c] = pr[c];
}

// ---------------------------------------------------------------------------
// Kernel 4: kNN over one candidate chunk. One wave per (32-query tile, chunk).
// D = A(16x4) x B(4x16) + C via V_WMMA_F32_16X16X4_F32:
//   A row  = ( x,  y,  z, 1 )            (queries)
//   B col  = (-2x,-2y,-2z,|q|^2)         (candidates)
//   C[m,n] = |p_m|^2 (row broadcast)
// => D[m,n] = full squared distance tile in ONE wmma.
// A layout (ISA 7.12.2): lanes0-15 hold K0(V0),K1(V1); lanes16-31 K2,K3.
// B layout assumed symmetric: V0 = rows {K0 | K2}, V1 = rows {K1 | K3}
// across lane halves -> exactly one global b64 load per lane per tile.
// ---------------------------------------------------------------------------
__global__ __launch_bounds__(32) void knn_chunk(const float* __restrict__ cand,
                                                const float* __restrict__ qpack,
                                                float* __restrict__ pdist,
                                                int* __restrict__ pidx) {
    __shared__ float4 qsv[QPW];            // staged query records
    __shared__ float  stage[QPW * 20];     // 32 rows x 16 (+4 pad) distance tile
    __shared__ float  bD[QPW * KNN];       // per-lane top-K distances
    __shared__ int    bI[QPW * KNN];       // per-lane top-K indices

    const int lane  = threadIdx.x;
    const int qt    = blockIdx.x;          // query tile (32 queries)
    const int ch    = blockIdx.y;          // candidate chunk
    const int b     = blockIdx.z;
    const int qbase = qt * QPW;

    qsv[lane] = ((const float4*)qpack)[(size_t)b * SGRP + qbase + lane];
    __syncthreads();

    const int hf = lane >> 4;              // which K-half this lane supplies
    const int lr = lane & 15;

    // A tiles for rows 0..15 (a0) and 16..31 (a1)
    v2f a0, a1;
    {
        float4 q0 = qsv[lr];
        a0[0] = hf ? q0.z : q0.x;
        a0[1] = hf ? 1.0f : q0.y;
        float4 q1 = qsv[16 + lr];
        a1[0] = hf ? q1.z : q1.x;
        a1[1] = hf ? 1.0f : q1.y;
    }
    // C accumulators: |p_m|^2 broadcast along columns (layout: VGPR v, lane
    // half h -> row m = v + 8*h)
    v8f c0, c1;
#pragma unroll
    for (int v = 0; v < 8; ++v) {
        c0[v] = qsv[v + 8 * hf].w;
        c1[v] = qsv[16 + v + 8 * hf].w;
    }

#pragma unroll
    for (int k = 0; k < KNN; ++k) { bD[lane * KNN + k] = 3e38f; bI[lane * KNN + k] = 0; }
    float worst = 3e38f; int wp = 0;

    const float2* cb = (const float2*)(cand + (size_t)b * NPTS * 4);
    const int n0 = ch * CHUNK_N;

    for (int tile = 0; tile < TILES_PER_CHUNK; ++tile) {
        const int nb = n0 + tile * CT;
        // per-lane B fragment: one 8B load (x2,y2) or (z2,|q|^2)
        const float2 bv = cb[(size_t)(nb + lr) * 2 + hf];
        if (tile + 1 < TILES_PER_CHUNK)
            __builtin_prefetch(&cb[(size_t)(nb + CT + lr) * 2 + hf], 0, 3); // WGP-scope
        v2f bvec; bvec[0] = bv.x; bvec[1] = bv.y;

        v8f d0 = __builtin_amdgcn_wmma_f32_16x16x4_f32(false, a0, false, bvec,
                                                       (short)0, c0, false, false);
        v8f d1 = __builtin_amdgcn_wmma_f32_16x16x4_f32(false, a1, false, bvec,
                                                       (short)0, c1, false, false);
        // scatter D tiles row-major into LDS (lane,VGPR v) -> (row v+8*hf, col lr)
#pragma unroll
        for (int v = 0; v < 8; ++v) {
            const int m = v + 8 * hf;
            stage[m * 20 + lr]        = d0[v];
            stage[(16 + m) * 20 + lr] = d1[v];
        }
        __syncthreads();
        // each lane owns one query row: replace-worst top-K selection
        const float* row = &stage[lane * 20];
#pragma unroll
        for (int j = 0; j < CT; ++j) {
            const float d = row[j];
            if (d < worst) {
                bD[lane * KNN + wp] = d;
                bI[lane * KNN + wp] = nb + j;
                float w2 = -1.0f; int p2 = 0;
#pragma unroll
                for (int k = 0; k < KNN; ++k) {
                    const float dk = bD[lane * KNN + k];
                    if (dk > w2) { w2 = dk; p2 = k; }
                }
                worst = w2; wp = p2;
            }
        }
        __syncthreads();
    }

    // sort ascending by distance (tie -> lower index), mirroring top_k order
    for (int i = 0; i < KNN; ++i) {
        int mk = i;
        for (int k = i + 1; k < KNN; ++k) {
            const float dk = bD[lane * KNN + k];
            const float dm = bD[lane * KNN + mk];
            if (dk < dm || (dk == dm && bI[lane * KNN + k] < bI[lane * KNN + mk])) mk = k;
        }
        const float td = bD[lane * KNN + i]; const int ti = bI[lane * KNN + i];
        bD[lane * KNN + i] = bD[lane * KNN + mk]; bI[lane * KNN + i] = bI[lane * KNN + mk];
        bD[lane * KNN + mk] = td; bI[lane * KNN + mk] = ti;
    }
    const size_t pb = (((size_t)b * SGRP + qbase + lane) * CHUNKS + ch) * KNN;
    for (int k = 0; k < KNN; ++k) {
        pdist[pb + k] = bD[lane * KNN + k];
        pidx[pb + k]  = bI[lane * KNN + k];
    }
}

// ---------------------------------------------------------------------------
// Kernel 5: merge CHUNKS sorted top-K lists per query into the final top-K.
// ---------------------------------------------------------------------------
__global__ __launch_bounds__(64) void knn_merge(const float* __restrict__ pdist,
                                                const int* __restrict__ pidx,
                                                int* __restrict__ fidx) {
    __shared__ float mD[64 * KNN];
    __shared__ int   mI[64 * KNN];
    const int tq = threadIdx.x;
    const int q  = blockIdx.x * 64 + tq;
    if (q >= BATCH * SGRP) return;

#pragma unroll
    for (int k = 0; k < KNN; ++k) { mD[tq * KNN + k] = 3e38f; mI[tq * KNN + k] = 0; }
    float worst = 3e38f; int wp = 0;
    const size_t base = (size_t)q * CHUNKS * KNN;

    for (int ch = 0; ch < CHUNKS; ++ch) {
        const size_t cb = base + (size_t)ch * KNN;
        for (int k = 0; k < KNN; ++k) {
            const float d = pdist[cb + k];
            if (d >= worst) break;                 // chunk list is sorted
            mD[tq * KNN + wp] = d;
            mI[tq * KNN + wp] = pidx[cb + k];
            float w2 = -1.0f; int p2 = 0;
#pragma unroll
            for (int j = 0; j < KNN; ++j) {
                const float dj = mD[tq * KNN + j];
                if (dj > w2) { w2 = dj; p2 = j; }
            }
            worst = w2; wp = p2;
        }
    }
    for (int i = 0; i < KNN; ++i) {
        int mk = i;
        for (int k = i + 1; k < KNN; ++k) {
            const float dk = mD[tq * KNN + k];
            const float dm = mD[tq * KNN + mk];
            if (dk < dm || (dk == dm && mI[tq * KNN + k] < mI[tq * KNN + mk])) mk = k;
        }
        const float td = mD[tq * KNN + i]; const int ti = mI[tq * KNN + i];
        mD[tq * KNN + i] = mD[tq * KNN + mk]; mI[tq * KNN + i] = mI[tq * KNN + mk];
        mD[tq * KNN + mk] = td; mI[tq * KNN + mk] = ti;
    }
    for (int k = 0; k < KNN; ++k) fidx[(size_t)q * KNN + k] = mI[tq * KNN + k];
}

// ---------------------------------------------------------------------------
// Kernel 6: gather grouped_xyz and grouped_points from final indices.
// ---------------------------------------------------------------------------
__global__ void out_gather(const float* __restrict__ xyz, const float* __restrict__ pts,
                           const int* __restrict__ fidx, float* __restrict__ out) {
    int g = blockIdx.x * blockDim.x + threadIdx.x;   // (b*SGRP+s)*KNN + k
    if (g >= BATCH * SGRP * KNN) return;
    const int bs = g / KNN;
    const int b  = bs / SGRP;
    const int i  = fidx[g];
    const float* xr = xyz + ((size_t)b * NPTS + i) * 3;
    float* gx = out + O_GXYZ + (size_t)g * 3;
    gx[0] = xr[0]; gx[1] = xr[1]; gx[2] = xr[2];
    const float4* pr = (const float4*)(pts + ((size_t)b * NPTS + i) * DCH);
    float4* po = (float4*)(out + O_GPTS + (size_t)g * DCH);
#pragma unroll
    for (int c = 0; c < DCH / 4; ++c) po[c] = pr[c];
}

// ---------------------------------------------------------------------------
extern "C" void kernel_launch(void* const* d_in, const int* in_sizes, int n_in,
                              void* d_out, int out_size, void* d_ws, size_t ws_size,
                              hipStream_t stream) {
    (void)in_sizes; (void)n_in; (void)out_size; (void)ws_size;
    const float* xyz = (const float*)d_in[0];
    const float* pts = (const float*)d_in[1];
    float* out = (float*)d_out;

    char* ws = (char*)d_ws;
    size_t off = 0;
    auto alloc = [&](size_t bytes) {
        size_t o = off;
        off += (bytes + 255) & ~(size_t)255;
        return o;
    };
    int*   fpsIdx = (int*)  (ws + alloc((size_t)BATCH * SGRP * sizeof(int)));
    float* cand   = (float*)(ws + alloc((size_t)BATCH * NPTS * 4 * sizeof(float)));
    float* qpack  = (float*)(ws + alloc((size_t)BATCH * SGRP * 4 * sizeof(float)));
    float* pdist  = (float*)(ws + alloc((size_t)BATCH * SGRP * CHUNKS * KNN * sizeof(float)));
    int*   pidx   = (int*)  (ws + alloc((size_t)BATCH * SGRP * CHUNKS * KNN * sizeof(int)));
    int*   fidx   = (int*)  (ws + alloc((size_t)BATCH * SGRP * KNN * sizeof(int)));

    prep_cand<<<(BATCH * NPTS + 255) / 256, 256, 0, stream>>>(xyz, cand);
    fps_kernel<<<BATCH, 1024, 0, stream>>>(xyz, fpsIdx);
    qgather<<<(BATCH * SGRP + 255) / 256, 256, 0, stream>>>(xyz, pts, fpsIdx, out, qpack);
    dim3 kg(SGRP / QPW, CHUNKS, BATCH);
    knn_chunk<<<kg, 32, 0, stream>>>(cand, qpack, pdist, pidx);
    knn_merge<<<(BATCH * SGRP) / 64, 64, 0, stream>>>(pdist, pidx, fidx);
    out_gather<<<(BATCH * SGRP * KNN + 255) / 256, 256, 0, stream>>>(xyz, pts, fidx, out);
}